// BPCA2D_44341242364811
// MI455X (gfx1250) — compile-verified
//
#include <hip/hip_runtime.h>

typedef float v2f __attribute__((ext_vector_type(2)));
typedef float v8f __attribute__((ext_vector_type(8)));

#define B_   16
#define C_   128
#define H_   96
#define W_   96
#define HO_  32
#define WO_  32
#define NROWS_ (C_ * HO_ * WO_)   // 131072 patch rows per batch

// ---------------------------------------------------------------------------
// Pass 1: padded Gram via V_WMMA_F32_16X16X4_F32.
// Feature vector per patch: [x0..x8, 1, 0,0,0,0,0,0]  (16-wide).
// Resulting 16x16 Gram: [0..8][0..8] raw second moments, row/col 9 = column
// sums, [9][9] = row count. Mean + covariance fall out of one accumulation.
// ---------------------------------------------------------------------------
__global__ __launch_bounds__(256) void bpca_gram_kernel(
    const float* __restrict__ x, float* __restrict__ gram) {
  __shared__ float tiles[8][32 * 16];   // per-wave 32 rows x 16 padded feats
  __shared__ float lgram[256];          // block-level 16x16 reduction buffer

  const int tid  = threadIdx.x;
  const int wave = tid >> 5;
  const int lane = tid & 31;
  const int b    = blockIdx.x >> 6;     // 64 blocks per batch
  const int blk  = blockIdx.x & 63;
  const int waveInBatch = blk * 8 + wave;   // 0..511; 8 strips each -> 4096

  lgram[tid] = 0.0f;
  float* tile = &tiles[wave][0];
  tile[lane * 16 + 9] = 1.0f;           // constant-1 feature (sums / count)
  #pragma unroll
  for (int j = 10; j < 16; ++j) tile[lane * 16 + j] = 0.0f;
  __syncthreads();

  v8f acc = {};                          // 16x16 f32 accumulator (8 VGPRs)
  for (int it = 0; it < 8; ++it) {
    const int strip = waveInBatch + it * 512;   // 0..4095, exact cover
    const int c  = strip >> 5;                  // /HO_
    const int ph = strip & 31;
    // lane = patch column pw; 3 contiguous 12B segments per lane (coalesced)
    const float* rowp =
        x + (((size_t)b * C_ + c) * H_ + (size_t)ph * 3) * W_ + lane * 3;
    float r[9];
    #pragma unroll
    for (int ky = 0; ky < 3; ++ky) {
      const float* p = rowp + ky * W_;
      r[ky * 3 + 0] = p[0];
      r[ky * 3 + 1] = p[1];
      r[ky * 3 + 2] = p[2];
    }
    // speculative prefetch of next strip (c += 16) -> global_prefetch_b8
    __builtin_prefetch(rowp + (size_t)16 * H_ * W_, 0, 1);

    #pragma unroll
    for (int j = 0; j < 9; ++j) tile[lane * 16 + j] = r[j];
    __syncthreads();   // uniform trip count across all 8 waves

    // f32 A-matrix 16x4 layout: lanes 0-15 {K0,K1}, lanes 16-31 {K2,K3}.
    // B = A^T chunk of the same data => pass the same register pair.
    const int m    = lane & 15;
    const int koff = (lane >= 16) ? 2 : 0;
    #pragma unroll
    for (int q = 0; q < 8; ++q) {       // 32 rows, K=4 per WMMA
      v2f a;
      a.x = tile[(q * 4 + koff + 0) * 16 + m];
      a.y = tile[(q * 4 + koff + 1) * 16 + m];
      acc = __builtin_amdgcn_wmma_f32_16x16x4_f32(
          /*neg_a=*/false, a, /*neg_b=*/false, a,
          /*c_mod=*/(short)0, acc, /*reuse_a=*/false, /*reuse_b=*/false);
    }
    __syncthreads();   // protect tile before next iteration's overwrite
  }

  // C/D layout: VGPR i -> lanes 0-15: (M=i, N=lane); lanes 16-31: (M=8+i).
  const int n  = lane & 15;
  const int mb = (lane < 16) ? 0 : 8;
  #pragma unroll
  for (int i = 0; i < 8; ++i)
    atomicAdd(&lgram[(mb + i) * 16 + n], acc[i]);   // ds_add_f32
  __syncthreads();
  atomicAdd(&gram[b * 256 + tid], lgram[tid]);      // global_atomic_add_f32
}

// ---------------------------------------------------------------------------
// Pass 2: per-batch 9x9 centered covariance -> top eigenvector via power
// iteration (deterministic fixed start + fixed iteration count). Tiny.
// vv[b][0..8] = v, vv[b][9] = dot(mu, v).
// ---------------------------------------------------------------------------
__global__ void bpca_eig_kernel(const float* __restrict__ gram,
                                float* __restrict__ vv) {
  const int b = threadIdx.x;
  if (b >= B_) return;
  const float* g = gram + b * 256;
  const float invN = 1.0f / (float)NROWS_;

  float mu[9];
  #pragma unroll
  for (int j = 0; j < 9; ++j) mu[j] = g[9 * 16 + j] * invN;

  float G[9][9];
  #pragma unroll
  for (int j = 0; j < 9; ++j)
    #pragma unroll
    for (int k = 0; k < 9; ++k)
      G[j][k] = g[j * 16 + k] - (float)NROWS_ * mu[j] * mu[k];

  float v[9];
  #pragma unroll
  for (int j = 0; j < 9; ++j) v[j] = 1.0f / 3.0f;
  for (int it = 0; it < 96; ++it) {
    float w[9];
    #pragma unroll
    for (int j = 0; j < 9; ++j) {
      float s = 0.0f;
      #pragma unroll
      for (int k = 0; k < 9; ++k) s += G[j][k] * v[k];
      w[j] = s;
    }
    float nrm = 0.0f;
    #pragma unroll
    for (int j = 0; j < 9; ++j) nrm += w[j] * w[j];
    const float r = rsqrtf(nrm + 1e-30f);
    #pragma unroll
    for (int j = 0; j < 9; ++j) v[j] = w[j] * r;
  }
  float md = 0.0f;
  #pragma unroll
  for (int j = 0; j < 9; ++j) md += mu[j] * v[j];
  #pragma unroll
  for (int j = 0; j < 9; ++j) vv[b * 16 + j] = v[j];
  vv[b * 16 + 9] = md;
}

// ---------------------------------------------------------------------------
// Pass 3: projection. Second sweep over x is L2-resident (75.5MB < 192MB L2).
// One wave per (b,c,ph) strip, lane = pw; fully coalesced loads and stores.
// ---------------------------------------------------------------------------
__global__ __launch_bounds__(256) void bpca_project_kernel(
    const float* __restrict__ x, const float* __restrict__ vv,
    float* __restrict__ out) {
  const int gid  = blockIdx.x * 8 + (threadIdx.x >> 5);  // strip 0..65535
  const int lane = threadIdx.x & 31;
  const int b    = gid >> 12;        // / (C_*HO_)
  const int rem  = gid & 4095;
  const int c    = rem >> 5;
  const int ph   = rem & 31;

  const float* vp = vv + b * 16;
  float v[9];
  #pragma unroll
  for (int j = 0; j < 9; ++j) v[j] = vp[j];   // uniform -> cache broadcast
  const float md = vp[9];

  const float* rowp =
      x + (((size_t)b * C_ + c) * H_ + (size_t)ph * 3) * W_ + lane * 3;
  float s = -md;
  #pragma unroll
  for (int ky = 0; ky < 3; ++ky) {
    const float* p = rowp + ky * W_;
    s += p[0] * v[ky * 3 + 0];
    s += p[1] * v[ky * 3 + 1];
    s += p[2] * v[ky * 3 + 2];
  }
  out[(size_t)gid * 32 + lane] = s;   // ((b*C+c)*HO+ph)*WO + pw
}

// ---------------------------------------------------------------------------
extern "C" void kernel_launch(void* const* d_in, const int* in_sizes, int n_in,
                              void* d_out, int out_size, void* d_ws,
                              size_t ws_size, hipStream_t stream) {
  (void)in_sizes; (void)n_in; (void)out_size; (void)ws_size;
  const float* x = (const float*)d_in[0];
  float* out  = (float*)d_out;
  float* gram = (float*)d_ws;          // B_*256 floats
  float* vv   = gram + B_ * 256;       // B_*16 floats

  // zero the Gram accumulators every launch (graph-capture safe)
  hipMemsetAsync(gram, 0, (size_t)(B_ * 256 + B_ * 16) * sizeof(float), stream);

  bpca_gram_kernel<<<B_ * 64, 256, 0, stream>>>(x, gram);
  bpca_eig_kernel<<<1, 32, 0, stream>>>(gram, vv);
  bpca_project_kernel<<<(B_ * C_ * HO_) / 8, 256, 0, stream>>>(x, vv, out);
}